// RegionAttentionMIL_26310969655772
// MI455X (gfx1250) — compile-verified
//
#include <hip/hip_runtime.h>
#include <math.h>

// Problem constants (match reference setup_inputs).
#define NB     8            // bins per axis
#define RGN    64           // NB*NB regions
#define D      1024         // feature dim
#define AA     128          // attention dim
#define HH     256          // classifier hidden dim
#define DTILE  256          // dims per segment-sum block
#define DTILES (D / DTILE)  // 4

typedef __attribute__((ext_vector_type(2))) float v2f;
typedef __attribute__((ext_vector_type(8))) float v8f;

// Monotone float<->uint key so atomicMin/Max on uints == float min/max (deterministic).
__device__ __forceinline__ unsigned fkey(float f) {
  unsigned u = __float_as_uint(f);
  return (u & 0x80000000u) ? ~u : (u | 0x80000000u);
}
__device__ __forceinline__ float funkey(unsigned k) {
  return (k & 0x80000000u) ? __uint_as_float(k & 0x7fffffffu) : __uint_as_float(~k);
}

// ---------------- K0: init workspace scalars ----------------
__global__ void k0_init(unsigned* mm, int* counts) {
  int t = threadIdx.x;
  if (t < 2) mm[t] = 0xffffffffu;   // min keys (x,y)
  else if (t < 4) mm[t] = 0u;       // max keys (x,y)
  if (t < RGN) counts[t] = 0;
}

// ---------------- K1: coords min/max ----------------
__global__ void __launch_bounds__(256)
k1_minmax(const float* __restrict__ coords, int N, unsigned* mm) {
  __shared__ unsigned smn0[256], smn1[256], smx0[256], smx1[256];
  int t = threadIdx.x;
  unsigned mn0 = 0xffffffffu, mn1 = 0xffffffffu, mx0 = 0u, mx1 = 0u;
  for (int i = blockIdx.x * blockDim.x + t; i < N; i += gridDim.x * blockDim.x) {
    unsigned k0 = fkey(coords[2 * i]);
    unsigned k1 = fkey(coords[2 * i + 1]);
    mn0 = min(mn0, k0); mx0 = max(mx0, k0);
    mn1 = min(mn1, k1); mx1 = max(mx1, k1);
  }
  smn0[t] = mn0; smn1[t] = mn1; smx0[t] = mx0; smx1[t] = mx1;
  __syncthreads();
  for (int s = 128; s > 0; s >>= 1) {
    if (t < s) {
      smn0[t] = min(smn0[t], smn0[t + s]); smn1[t] = min(smn1[t], smn1[t + s]);
      smx0[t] = max(smx0[t], smx0[t + s]); smx1[t] = max(smx1[t], smx1[t + s]);
    }
    __syncthreads();
  }
  if (t == 0) {
    atomicMin(&mm[0], smn0[0]); atomicMin(&mm[1], smn1[0]);
    atomicMax(&mm[2], smx0[0]); atomicMax(&mm[3], smx1[0]);
  }
}

// ---------------- K2: region ids + counts ----------------
__global__ void __launch_bounds__(256)
k2_rid(const float* __restrict__ coords, int N, const unsigned* __restrict__ mm,
       int* __restrict__ rid, int* __restrict__ counts) {
  int i = blockIdx.x * blockDim.x + threadIdx.x;
  if (i >= N) return;
  float lo0 = funkey(mm[0]), lo1 = funkey(mm[1]);
  float hi0 = funkey(mm[2]), hi1 = funkey(mm[3]);
  float sp0 = fmaxf(hi0 - lo0, 1.0f), sp1 = fmaxf(hi1 - lo1, 1.0f);
  float c0 = coords[2 * i], c1 = coords[2 * i + 1];
  int b0 = (int)((c0 - lo0) / sp0 * (float)NB); b0 = min(max(b0, 0), NB - 1);
  int b1 = (int)((c1 - lo1) / sp1 * (float)NB); b1 = min(max(b1, 0), NB - 1);
  int r = b1 * NB + b0;
  rid[i] = r;
  atomicAdd(&counts[r], 1);   // integer: order-independent -> deterministic
}

// ---------------- K3: deterministic segment-sum (the 410 MB streaming pass) ----
// Lane t owns dim column dBase+t; LDS slot [region][t] has a single writer lane,
// so ds_add_f32 accumulation order is program order -> bitwise deterministic.
__global__ void __launch_bounds__(256)
k3_segsum(const float* __restrict__ x, const int* __restrict__ rid, int N,
          int rowsPerChunk, float* __restrict__ partial) {
  extern __shared__ float lds[];              // RGN * DTILE floats = 64 KB
  const int t = threadIdx.x;
  const int dBase = blockIdx.x * DTILE;
  const int chunk = blockIdx.y;
#pragma unroll
  for (int rg = 0; rg < RGN; ++rg) lds[rg * DTILE + t] = 0.0f;
  __syncthreads();

  int r0 = chunk * rowsPerChunk;
  int r1 = min(r0 + rowsPerChunk, N);
  const float* xp = x + (size_t)r0 * D + dBase + t;
  int row = r0;
  for (; row + 4 <= r1; row += 4, xp += 4 * (size_t)D) {
    int rg0 = rid[row], rg1 = rid[row + 1], rg2 = rid[row + 2], rg3 = rid[row + 3];
    float v0 = __builtin_nontemporal_load(xp);
    float v1 = __builtin_nontemporal_load(xp + D);
    float v2 = __builtin_nontemporal_load(xp + 2 * D);
    float v3 = __builtin_nontemporal_load(xp + 3 * D);
    atomicAdd(&lds[rg0 * DTILE + t], v0);    // ds_add_f32 (no return)
    atomicAdd(&lds[rg1 * DTILE + t], v1);
    atomicAdd(&lds[rg2 * DTILE + t], v2);
    atomicAdd(&lds[rg3 * DTILE + t], v3);
  }
  for (; row < r1; ++row, xp += D) {
    int rg = rid[row];
    float v = __builtin_nontemporal_load(xp);
    atomicAdd(&lds[rg * DTILE + t], v);
  }
  __syncthreads();

  float* pp = partial + ((size_t)chunk * RGN) * D + dBase + t;
#pragma unroll
  for (int rg = 0; rg < RGN; ++rg) pp[(size_t)rg * D] = lds[rg * DTILE + t];
}

// ---------------- K4: fixed-order chunk reduction -> region means ----------------
__global__ void __launch_bounds__(256)
k4_reduce(const float* __restrict__ partial, const int* __restrict__ counts,
          int nchunk, float* __restrict__ rmean) {
  int idx = blockIdx.x * blockDim.x + threadIdx.x;   // [0, RGN*D)
  int rg = idx >> 10;
  int dim = idx & (D - 1);
  float s = 0.0f;
  for (int c = 0; c < nchunk; ++c)                   // fixed order -> deterministic
    s += partial[((size_t)c * RGN + rg) * D + dim];
  float cnt = (float)counts[rg];
  rmean[idx] = s / fmaxf(cnt, 1.0f);
}

// ---------------- K5: r @ U_w.T and r @ V_w.T via V_WMMA_F32_16X16X4_F32 -------
// One wave per 16x16 output tile. A(16x4) f32 layout: lanes 0-15 -> M=lane,
// VGPR{0,1}=K{0,1}; lanes 16-31 -> K{2,3}. B(4x16): lanes hold N, half-waves hold
// K{0,1}/K{2,3}. Both fragments are contiguous float2 loads.
__global__ void __launch_bounds__(32)
k5_gates_wmma(const float* __restrict__ rmean,
              const float* __restrict__ Uw, const float* __restrict__ Vw,
              float* __restrict__ Zu, float* __restrict__ Zv) {
  const int lane = threadIdx.x;
  const int half = lane >> 4;
  const int l = lane & 15;
  const int m0 = (blockIdx.x >> 3) * 16;   // 4 M-tiles (64 regions)
  const int n0 = (blockIdx.x & 7) * 16;    // 8 N-tiles (128 attn dims)
  const float* W = blockIdx.y ? Vw : Uw;
  float* Z = blockIdx.y ? Zv : Zu;

  const float* ap = rmean + (size_t)(m0 + l) * D + 2 * half;  // A[m][k..k+1]
  const float* bp = W + (size_t)(n0 + l) * D + 2 * half;      // B[k][n] = W[n][k]
  v8f acc = {0.f, 0.f, 0.f, 0.f, 0.f, 0.f, 0.f, 0.f};
#pragma unroll 8
  for (int k = 0; k < D; k += 4) {
    v2f a = *(const v2f*)(ap + k);
    v2f b = *(const v2f*)(bp + k);
    acc = __builtin_amdgcn_wmma_f32_16x16x4_f32(
        /*neg_a=*/false, a, /*neg_b=*/false, b,
        /*c_mod=*/(short)0, acc, /*reuse_a=*/false, /*reuse_b=*/false);
  }
  // C/D layout: VGPR v -> M = v + 8*half, N = lane&15.
#pragma unroll
  for (int v = 0; v < 8; ++v)
    Z[(size_t)(m0 + v + 8 * half) * AA + n0 + l] = acc[v];
}

// ---------------- K6: gating + masked softmax + slide + MLP head ----------------
__global__ void __launch_bounds__(256)
k6_final(const float* __restrict__ Zu, const float* __restrict__ Zv,
         const float* __restrict__ Ub, const float* __restrict__ Vb,
         const float* __restrict__ ww, const float* __restrict__ rmean,
         const int* __restrict__ counts,
         const float* __restrict__ c1w, const float* __restrict__ c1b,
         const float* __restrict__ c2w, const float* __restrict__ c2b,
         float* __restrict__ out) {
  __shared__ float scp[RGN][4];
  __shared__ float attn[RGN];
  __shared__ float slide[D];
  __shared__ float hbuf[HH];
  const int t = threadIdx.x;

  // Gated scores: 4 threads per region, each covers 32 of 128 attn dims.
  {
    int m = t >> 2, p = t & 3;
    float s = 0.0f;
    for (int a = p * 32; a < p * 32 + 32; ++a) {
      float zu = Zu[m * AA + a] + Ub[a];
      float zv = Zv[m * AA + a] + Vb[a];
      float g = tanhf(zu) * (1.0f / (1.0f + expf(-zv)));
      s += g * ww[a];
    }
    scp[m][p] = s;
  }
  __syncthreads();
  if (t < RGN) {
    float s = scp[t][0] + scp[t][1] + scp[t][2] + scp[t][3];   // fixed order
    attn[t] = (counts[t] > 0) ? s : -__builtin_inff();
  }
  __syncthreads();
  if (t == 0) {                                                // serial softmax: deterministic
    float mx = -__builtin_inff();
    for (int m = 0; m < RGN; ++m) mx = fmaxf(mx, attn[m]);
    float sum = 0.0f;
    for (int m = 0; m < RGN; ++m) { float e = expf(attn[m] - mx); attn[m] = e; sum += e; }
    float inv = 1.0f / sum;
    for (int m = 0; m < RGN; ++m) attn[m] *= inv;
  }
  __syncthreads();
  // slide[d] = sum_m attn[m] * r[m][d]
  for (int d = t; d < D; d += 256) {
    float s = 0.0f;
    for (int m = 0; m < RGN; ++m) s += attn[m] * rmean[m * D + d];
    slide[d] = s;
    out[1 + d] = s;
  }
  if (t < RGN) out[1 + D + t] = attn[t];
  __syncthreads();
  // h = relu(slide @ c1_w.T + c1_b); one thread per hidden unit.
  {
    float acc = c1b[t];
    for (int k = 0; k < D; ++k) acc += slide[k] * c1w[(size_t)t * D + k];
    hbuf[t] = fmaxf(acc, 0.0f);
  }
  __syncthreads();
  if (t == 0) {
    float lg = c2b[0];
    for (int j = 0; j < HH; ++j) lg += hbuf[j] * c2w[j];
    out[0] = lg;
  }
}

// ---------------- host launch ----------------
extern "C" void kernel_launch(void* const* d_in, const int* in_sizes, int n_in,
                              void* d_out, int out_size, void* d_ws, size_t ws_size,
                              hipStream_t stream) {
  const float* x      = (const float*)d_in[0];
  const float* coords = (const float*)d_in[1];
  const float* Uw     = (const float*)d_in[2];
  const float* Ub     = (const float*)d_in[3];
  const float* Vw     = (const float*)d_in[4];
  const float* Vb     = (const float*)d_in[5];
  const float* ww     = (const float*)d_in[6];
  const float* c1w    = (const float*)d_in[7];
  const float* c1b    = (const float*)d_in[8];
  const float* c2w    = (const float*)d_in[9];
  const float* c2b    = (const float*)d_in[10];
  float* out = (float*)d_out;
  const int N = in_sizes[1] / 2;

  // Workspace carve-up (256-B aligned slabs).
  char* ws = (char*)d_ws;
  size_t off = 0;
  auto take = [&](size_t bytes) -> void* {
    void* p = ws + off;
    off = (off + bytes + 255) & ~(size_t)255;
    return p;
  };
  unsigned* mm  = (unsigned*)take(4 * sizeof(unsigned));
  int* counts   = (int*)take(RGN * sizeof(int));
  int* rid      = (int*)take((size_t)N * sizeof(int));
  float* rmean  = (float*)take((size_t)RGN * D * sizeof(float));
  float* Zu     = (float*)take((size_t)RGN * AA * sizeof(float));
  float* Zv     = (float*)take((size_t)RGN * AA * sizeof(float));

  const size_t chunkBytes = (size_t)RGN * D * sizeof(float);   // 256 KB per chunk
  size_t avail = (ws_size > off) ? (ws_size - off) : 0;
  int nchunk = (int)(avail / chunkBytes);
  if (nchunk < 1) nchunk = 1;
  if (nchunk > 256) nchunk = 256;       // 1024 blocks in K3, 64 MB partials max
  float* partial = (float*)(ws + off);
  int rowsPerChunk = (N + nchunk - 1) / nchunk;

  k0_init<<<1, 256, 0, stream>>>(mm, counts);
  k1_minmax<<<192, 256, 0, stream>>>(coords, N, mm);
  k2_rid<<<(N + 255) / 256, 256, 0, stream>>>(coords, N, mm, rid, counts);
  k3_segsum<<<dim3(DTILES, nchunk), 256, RGN * DTILE * sizeof(float), stream>>>(
      x, rid, N, rowsPerChunk, partial);
  k4_reduce<<<(RGN * D) / 256, 256, 0, stream>>>(partial, counts, nchunk, rmean);
  k5_gates_wmma<<<dim3(32, 2), 32, 0, stream>>>(rmean, Uw, Vw, Zu, Zv);
  k6_final<<<1, 256, 0, stream>>>(Zu, Zv, Ub, Vb, ww, rmean, counts,
                                  c1w, c1b, c2w, c2b, out);
}